// enc_dec_attention_40235253629311
// MI455X (gfx1250) — compile-verified
//
#include <hip/hip_runtime.h>

typedef __attribute__((ext_vector_type(2))) float v2f;
typedef __attribute__((ext_vector_type(8))) float v8f;

#define BDIM 8
#define MDIM 256
#define NDIM 256
#define CDIM 128
#define LDP  132   // padded LDS row stride (floats): 132 % 64 == 4 -> conflict-free b64 reads

__device__ __forceinline__ v8f wmma_f32_4(v2f a, v2f b, v8f c) {
    // D = A(16x4 f32) x B(4x16 f32) + C(16x16 f32)
    return __builtin_amdgcn_wmma_f32_16x16x4_f32(false, a, false, b, (short)0, c, false, false);
}

// ---------------------------------------------------------------------------
// Generic 16-row x 128-col GEMM-with-bias: out = X @ W.T + bias
// one wave per block, rows = blockIdx.x*16 .. +15
// ---------------------------------------------------------------------------
__global__ void linear_rowtile_kernel(const float* __restrict__ X,
                                      const float* __restrict__ W,
                                      const float* __restrict__ bias,
                                      float* __restrict__ out) {
    const int l   = threadIdx.x;       // 0..31
    const int lo  = l & 15;
    const int hi  = l >> 4;
    const int off = hi * 2;
    const int r0  = blockIdx.x * 16;

    const float* xrow = X + (size_t)(r0 + lo) * CDIM + off;
    for (int jt = 0; jt < 8; ++jt) {
        v8f acc = {};
        const float* wrow = W + (size_t)(jt * 16 + lo) * CDIM + off;
        #pragma unroll
        for (int kk = 0; kk < 32; ++kk) {
            v2f a = *(const v2f*)(xrow + 4 * kk);   // A[m=lo][k0+off..+1]
            v2f b = *(const v2f*)(wrow + 4 * kk);   // B[k][n=lo] = W[n][k]
            acc = wmma_f32_4(a, b, acc);
        }
        const int   col = jt * 16 + lo;
        const float bv  = bias[col];
        const int   rb  = r0 + hi * 8;
        #pragma unroll
        for (int r = 0; r < 8; ++r)
            out[(size_t)(rb + r) * CDIM + col] = acc[r] + bv;
    }
}

// ---------------------------------------------------------------------------
// Fused: e = adj@Wa.T+ba ; attn = q*k*e/sqrt(C) (-> ws) ; adj_out = attn@We.T+be
// One block per (b,m). 8 waves; each wave owns two 16-row n-tiles.
// ---------------------------------------------------------------------------
__global__ __launch_bounds__(256) void fused_edge_kernel(
    const float* __restrict__ adj,     // [B,M,N,C]
    const float* __restrict__ Wa, const float* __restrict__ ba,
    const float* __restrict__ We, const float* __restrict__ be,
    const float* __restrict__ qws,     // [B*M, C]
    const float* __restrict__ kws,     // [B*N, C]
    float* __restrict__ attn_ws,       // [B,M,N,C]
    float* __restrict__ adj_out)       // [B,M,N,C]
{
    extern __shared__ float smem[];
    float* sWa   = smem;                       // 128*LDP
    float* sWe   = sWa + CDIM * LDP;           // 128*LDP
    float* sAttn = sWe + CDIM * LDP;           // 8 waves * 16*LDP
    float* sQ    = sAttn + 8 * 16 * LDP;       // 128
    float* sBa   = sQ + CDIM;                  // 128
    float* sBe   = sBa + CDIM;                 // 128

    const int tid = threadIdx.x;
    const int bm  = blockIdx.x;     // b*M + m
    const int b   = bm >> 8;

    // cooperative stage of Wa, We into padded LDS (float4 chunks)
    for (int idx = tid; idx < CDIM * 32; idx += 256) {
        const int row = idx >> 5;
        const int cv  = (idx & 31) * 4;
        const float4 wa = *(const float4*)(Wa + (size_t)idx * 4);
        const float4 we = *(const float4*)(We + (size_t)idx * 4);
        float* pa = sWa + row * LDP + cv;
        float* pe = sWe + row * LDP + cv;
        pa[0] = wa.x; pa[1] = wa.y; pa[2] = wa.z; pa[3] = wa.w;
        pe[0] = we.x; pe[1] = we.y; pe[2] = we.z; pe[3] = we.w;
    }
    if (tid < CDIM) {
        sQ[tid]  = qws[(size_t)bm * CDIM + tid];
        sBa[tid] = ba[tid];
        sBe[tid] = be[tid];
    }
    __syncthreads();

    const int wave = tid >> 5;
    const int l    = tid & 31;
    const int lo   = l & 15;
    const int hi   = l >> 4;
    const int off  = hi * 2;
    float* myAttn = sAttn + wave * 16 * LDP;
    const float scale = 0.088388347648318447f;   // 1/sqrt(128)

    for (int tt = 0; tt < 2; ++tt) {
        const int    n0      = (wave + tt * 8) * 16;
        const size_t rowBase = (size_t)bm * NDIM + n0;

        // preload adj-tile A fragments (whole 16x128 tile across the wave)
        v2f aF[32];
        {
            const float* arow = adj + (rowBase + lo) * CDIM + off;
            #pragma unroll
            for (int kk = 0; kk < 32; ++kk)
                aF[kk] = *(const v2f*)(arow + 4 * kk);
        }

        // GEMM1 (e) + elementwise attn, 8 column tiles
        for (int jt = 0; jt < 8; ++jt) {
            v8f acc = {};
            const float* wrow = sWa + (jt * 16 + lo) * LDP + off;
            #pragma unroll
            for (int kk = 0; kk < 32; ++kk) {
                v2f bF = *(const v2f*)(wrow + 4 * kk);
                acc = wmma_f32_4(aF[kk], bF, acc);
            }
            const int   c   = jt * 16 + lo;
            const float qv  = sQ[c] * scale;
            const float bav = sBa[c];
            #pragma unroll
            for (int r = 0; r < 8; ++r) {
                const int   nn = r + hi * 8;
                const float kv = kws[((size_t)b * NDIM + n0 + nn) * CDIM + c];
                const float at = qv * kv * (acc[r] + bav);
                attn_ws[(rowBase + nn) * CDIM + c] = at;
                myAttn[nn * LDP + c] = at;
            }
        }
        __syncthreads();   // publish attn tile (cross-lane LDS RAW)

        // reload A fragments = attn tile (now from LDS)
        {
            const float* arow = myAttn + lo * LDP + off;
            #pragma unroll
            for (int kk = 0; kk < 32; ++kk)
                aF[kk] = *(const v2f*)(arow + 4 * kk);
        }

        // GEMM2: adj_out = attn @ We.T + be
        for (int jt = 0; jt < 8; ++jt) {
            v8f acc = {};
            const float* wrow = sWe + (jt * 16 + lo) * LDP + off;
            #pragma unroll
            for (int kk = 0; kk < 32; ++kk) {
                v2f bF = *(const v2f*)(wrow + 4 * kk);
                acc = wmma_f32_4(aF[kk], bF, acc);
            }
            const int   c   = jt * 16 + lo;
            const float bev = sBe[c];
            #pragma unroll
            for (int r = 0; r < 8; ++r) {
                const int nn = r + hi * 8;
                adj_out[(rowBase + nn) * CDIM + c] = acc[r] + bev;
            }
        }
        __syncthreads();   // done reading myAttn before next tile overwrites it
    }
}

// ---------------------------------------------------------------------------
// Online softmax over n + value-weighted sum: node[bm,c]
// ---------------------------------------------------------------------------
__global__ void softmax_node_kernel(const float* __restrict__ attn_ws,
                                    const float* __restrict__ vws,
                                    float* __restrict__ node_ws) {
    const int bm = blockIdx.x;
    const int c  = threadIdx.x;      // 0..127
    const int b  = bm >> 8;
    const float* ap = attn_ws + (size_t)bm * NDIM * CDIM + c;
    const float* vp = vws + (size_t)b * NDIM * CDIM + c;
    float m = -3.4e38f, s = 0.f, acc = 0.f;
    for (int n = 0; n < NDIM; ++n) {
        const float x  = ap[(size_t)n * CDIM];
        const float vv = vp[(size_t)n * CDIM];
        const float nm = fmaxf(m, x);
        const float cf = __expf(m - nm);
        const float e  = __expf(x - nm);
        s   = s * cf + e;
        acc = acc * cf + e * vv;
        m = nm;
    }
    node_ws[(size_t)bm * CDIM + c] = acc / s;
}

// ---------------------------------------------------------------------------
extern "C" void kernel_launch(void* const* d_in, const int* in_sizes, int n_in,
                              void* d_out, int out_size, void* d_ws, size_t ws_size,
                              hipStream_t stream) {
    const float* mol_annot = (const float*)d_in[0];
    const float* mol_adj   = (const float*)d_in[1];
    const float* prot      = (const float*)d_in[2];
    const float* Wq = (const float*)d_in[3];  const float* bq = (const float*)d_in[4];
    const float* Wk = (const float*)d_in[5];  const float* bk = (const float*)d_in[6];
    const float* Wv = (const float*)d_in[7];  const float* bv = (const float*)d_in[8];
    const float* Wa = (const float*)d_in[9];  const float* ba = (const float*)d_in[10];
    const float* Wn = (const float*)d_in[11]; const float* bn = (const float*)d_in[12];
    const float* We = (const float*)d_in[13]; const float* be = (const float*)d_in[14];

    const size_t BMC = (size_t)BDIM * MDIM * CDIM;  // 262144
    float* qws    = (float*)d_ws;
    float* kws    = qws + BMC;
    float* vws    = kws + BMC;
    float* nodews = vws + BMC;
    float* attnws = nodews + BMC;                    // 67,108,864 floats

    float* node_out = (float*)d_out;                 // [B,M,C]
    float* adj_out  = node_out + BMC;                // [B,M,N,C]

    const int rowTiles = (BDIM * MDIM) / 16;         // 128

    // q, k, v projections (16-row WMMA tiles)
    linear_rowtile_kernel<<<rowTiles, 32, 0, stream>>>(mol_annot, Wq, bq, qws);
    linear_rowtile_kernel<<<rowTiles, 32, 0, stream>>>(prot,      Wk, bk, kws);
    linear_rowtile_kernel<<<rowTiles, 32, 0, stream>>>(mol_annot, Wv, bv, vws);

    // fused edge path: GEMM1 -> elementwise attn -> GEMM2
    const size_t lds = (size_t)(2 * CDIM * LDP + 8 * 16 * LDP + 3 * CDIM) * sizeof(float);
    fused_edge_kernel<<<BDIM * MDIM, 256, lds, stream>>>(mol_adj, Wa, ba, We, be,
                                                         qws, kws, attnws, adj_out);

    // node path: online softmax over n + weighted v, then final projection
    softmax_node_kernel<<<BDIM * MDIM, CDIM, 0, stream>>>(attnws, vws, nodews);
    linear_rowtile_kernel<<<rowTiles, 32, 0, stream>>>(nodews, Wn, bn, node_out);
}